// ManualCFConv_61418032333201
// MI455X (gfx1250) — compile-verified
//
#include <hip/hip_runtime.h>
#include <hip/hip_bf16.h>
#include <stddef.h>

// CDNA5 / gfx1250, wave32. FP32 WMMA (V_WMMA_F32_16X16X4_F32) edge-MLP +
// segment-softmax scatter. Memory-bound (edge_attr stream ~102MB/pass at
// 23.3 TB/s); fp32 matrix path keeps reference numerics at zero roofline cost.
// Weights are pre-swizzled in LDS into WMMA B-fragment order so every
// fragment is a single contiguous ds_load_b64 (no VGPR re-packing movs,
// conflict-free: a wave reads 256B = all 64 banks exactly once).

typedef __attribute__((ext_vector_type(2))) float v2f;
typedef __attribute__((ext_vector_type(8))) float v8f;

#define D_EDGE 32
#define D_FEAT 64
#define WPB 4            // waves per block
#define EPW 16           // edges per wave tile (WMMA M)
#define EPB (WPB * EPW)  // edges per block
#define HS  (D_FEAT + 4) // sH row stride: 8B-aligned, bank shift 4/row

__device__ __forceinline__ float fast_tanh(float x) {
  // tanh(x) = 1 - 2/(exp(2x)+1); exact saturation at +-1, no NaN at +-inf
  float e = __expf(2.0f * x);
  return 1.0f - 2.0f * __builtin_amdgcn_rcpf(e + 1.0f);
}

__global__ void init_kernel(float* out, float* smax, float* denom, int n) {
  int i = blockIdx.x * blockDim.x + threadIdx.x;
  if (i < n * D_FEAT) out[i] = 0.0f;
  if (i < n) {
    ((unsigned*)smax)[i] = 0xFF800000u; // -inf
    denom[i] = 0.0f;
  }
}

__global__ void segmax_kernel(const float* __restrict__ scores,
                              const long long* __restrict__ tgt,
                              float* smax, int E) {
  int e = blockIdx.x * blockDim.x + threadIdx.x;
  if (e >= E) return;
  int t = (int)tgt[e];
  float s = scores[e];
  // monotonic-bits float max: int-max for >=0, uint-min for <0
  if (s >= 0.0f) atomicMax((int*)(smax + t), __float_as_int(s));
  else           atomicMin((unsigned*)(smax + t), __float_as_uint(s));
}

__global__ void segsum_kernel(const float* __restrict__ scores,
                              const long long* __restrict__ tgt,
                              const float* __restrict__ smax,
                              float* denom, int E) {
  int e = blockIdx.x * blockDim.x + threadIdx.x;
  if (e >= E) return;
  int t = (int)tgt[e];
  atomicAdd(denom + t, __expf(scores[e] - smax[t]));
}

// B-fragment swizzle: element W[k][n] (row-major k*64+n) lands at float index
//   pair = ((k>>2)*4 + (n>>4))*32 + ((k>>1)&1)*16 + (n&15)
//   addr = pair*2 + (k&1)
// so the fragment for (kc, nt) is read by lane L as the v2f at
//   sW[ ((kc*4 + nt)*32 + L) * 2 ]
__device__ __forceinline__ int bswz(int k, int n) {
  return ((((k >> 2) * 4 + (n >> 4)) * 32 + (((k >> 1) & 1) * 16) + (n & 15)) << 1)
         + (k & 1);
}

__global__ __launch_bounds__(128)
void edge_mlp_kernel(const float* __restrict__ x,
                     const long long* __restrict__ srcIdx,
                     const long long* __restrict__ tgtIdx,
                     const float* __restrict__ edge_attr,
                     const float* __restrict__ W1,
                     const float* __restrict__ b1,
                     const float* __restrict__ W2,
                     const float* __restrict__ b2,
                     const float* __restrict__ av,
                     float* scores,                      // phase 0 output
                     const float* __restrict__ smax,     // phase 1 inputs
                     const float* __restrict__ denom,
                     float* out,                         // phase 1 atomic dst
                     float* attn_out,                    // phase 1 output
                     int E, int phase) {
  __shared__ float sW1[D_EDGE * D_FEAT];   // B-fragment swizzled, 8KB
  __shared__ float sW2[D_FEAT * D_FEAT];   // B-fragment swizzled, 16KB
  __shared__ float sB1[D_FEAT];
  __shared__ float sB2[D_FEAT];
  __shared__ float sAV[D_FEAT];
  __shared__ float sH[WPB][EPW][HS];       // per-wave D->A relayout scratch
  __shared__ int   sSrc[WPB][EPW];
  __shared__ int   sTgt[WPB][EPW];

  int tid = threadIdx.x;
  for (int i = tid; i < D_EDGE * D_FEAT; i += 128)
    sW1[bswz(i >> 6, i & 63)] = W1[i];
  for (int i = tid; i < D_FEAT * D_FEAT; i += 128)
    sW2[bswz(i >> 6, i & 63)] = W2[i];
  if (tid < D_FEAT) { sB1[tid] = b1[tid]; sB2[tid] = b2[tid]; sAV[tid] = av[tid]; }
  __syncthreads();

  int wave = tid >> 5;
  int lane = tid & 31;
  int e0 = (blockIdx.x * WPB + wave) * EPW;
  if (e0 + EPW > E) return; // wave-uniform: EXEC all-ones inside (E % 16 == 0)

  if (lane < EPW) {
    sSrc[wave][lane] = (int)srcIdx[e0 + lane];
    sTgt[wave][lane] = (int)tgtIdx[e0 + lane];
  }

  int row = lane & 15;   // A-frag row / B-frag,D col
  int hi  = lane >> 4;   // half-wave select
  int col = row;

  // ---- A fragments of edge_attr [16 x 32] (fp32 16x16x4 layout) ----
  v2f afrag[8];
  const float* ea = edge_attr + (size_t)e0 * D_EDGE;
  #pragma unroll
  for (int kc = 0; kc < 8; ++kc) {
    int k = 4 * kc + 2 * hi;
    afrag[kc] = *(const v2f*)(ea + row * D_EDGE + k); // global_load_b64
  }

  // ---- GEMM1: h = tanh(edge_attr @ W1 + b1) -> LDS (A layout for GEMM2) ----
  #pragma unroll
  for (int nt = 0; nt < 4; ++nt) {
    v8f acc = {0, 0, 0, 0, 0, 0, 0, 0};
    #pragma unroll
    for (int kc = 0; kc < 8; ++kc) {
      v2f b = *(const v2f*)&sW1[(((kc * 4 + nt) * 32) + lane) << 1]; // ds_load_b64
      acc = __builtin_amdgcn_wmma_f32_16x16x4_f32(false, afrag[kc], false, b,
                                                  (short)0, acc, false, false);
    }
    int n0 = nt * 16;
    float bias = sB1[n0 + col];
    #pragma unroll
    for (int r = 0; r < 8; ++r)
      sH[wave][r + 8 * hi][n0 + col] = fast_tanh(acc[r] + bias);
  }
  // same-wave DS write->read is in-order; compiler inserts s_wait_dscnt

  // ---- GEMM2: ew = tanh(h @ W2 + b2), kept in D layout ----
  v8f ew[4];
  #pragma unroll
  for (int nt = 0; nt < 4; ++nt) {
    v8f acc = {0, 0, 0, 0, 0, 0, 0, 0};
    #pragma unroll
    for (int kc = 0; kc < 16; ++kc) {
      int k = 4 * kc + 2 * hi;
      v2f a = *(const v2f*)&sH[wave][row][k];                        // ds_load_b64
      v2f b = *(const v2f*)&sW2[(((kc * 4 + nt) * 32) + lane) << 1]; // ds_load_b64
      acc = __builtin_amdgcn_wmma_f32_16x16x4_f32(false, a, false, b,
                                                  (short)0, acc, false, false);
    }
    int n0 = nt * 16;
    float bias = sB2[n0 + col];
    #pragma unroll
    for (int r = 0; r < 8; ++r) acc[r] = fast_tanh(acc[r] + bias);
    ew[nt] = acc;
  }

  // ---- messages = x[src] * ew ; scores = messages . attn_vec ----
  float sc[8] = {0, 0, 0, 0, 0, 0, 0, 0};
  #pragma unroll
  for (int nt = 0; nt < 4; ++nt) {
    int f = nt * 16 + col;
    float a = sAV[f];
    #pragma unroll
    for (int r = 0; r < 8; ++r) {
      int m = r + 8 * hi;
      int s = sSrc[wave][m];
      float mv = x[(size_t)s * D_FEAT + f] * ew[nt][r]; // gather: x resident in L2
      ew[nt][r] = mv;
      sc[r] += mv * a;
    }
  }
  // reduce over the 16 lanes of each half-wave (feature dimension)
  #pragma unroll
  for (int r = 0; r < 8; ++r) {
    float v = sc[r];
    v += __shfl_xor(v, 1, 32);
    v += __shfl_xor(v, 2, 32);
    v += __shfl_xor(v, 4, 32);
    v += __shfl_xor(v, 8, 32);
    sc[r] = v;
  }

  if (phase == 0) {
    if (col == 0) {
      #pragma unroll
      for (int r = 0; r < 8; ++r) scores[e0 + r + 8 * hi] = sc[r];
    }
    return;
  }

  // ---- phase 1: softmax-normalize + scatter-add ----
  float attn[8];
  #pragma unroll
  for (int r = 0; r < 8; ++r) {
    int t = sTgt[wave][r + 8 * hi];
    attn[r] = __expf(sc[r] - smax[t]) / denom[t];
  }
  if (col == 0) {
    #pragma unroll
    for (int r = 0; r < 8; ++r) attn_out[e0 + r + 8 * hi] = attn[r];
  }
  #pragma unroll
  for (int r = 0; r < 8; ++r) {
    int t = sTgt[wave][r + 8 * hi];
    size_t base = (size_t)t * D_FEAT + col;
    #pragma unroll
    for (int nt = 0; nt < 4; ++nt)
      atomicAdd(out + base + nt * 16, ew[nt][r] * attn[r]); // global_atomic_add_f32
  }
}

extern "C" void kernel_launch(void* const* d_in, const int* in_sizes, int n_in,
                              void* d_out, int out_size, void* d_ws, size_t ws_size,
                              hipStream_t stream) {
  (void)n_in; (void)out_size; (void)ws_size;
  const float*     x         = (const float*)d_in[0];
  const long long* ei        = (const long long*)d_in[1]; // int64 [2, E]
  const float*     edge_attr = (const float*)d_in[2];
  const float*     W1        = (const float*)d_in[3];
  const float*     b1        = (const float*)d_in[4];
  const float*     W2        = (const float*)d_in[5];
  const float*     b2        = (const float*)d_in[6];
  const float*     av        = (const float*)d_in[7];

  int n = in_sizes[0] / D_FEAT;
  int E = in_sizes[1] / 2;
  const long long* src = ei;
  const long long* tgt = ei + E;

  float* out      = (float*)d_out;
  float* attn_out = out + (size_t)n * D_FEAT;

  // workspace: scores[E], smax[n], denom[n]  (~3.6 MB)
  float* scores = (float*)d_ws;
  float* smax   = scores + E;
  float* denom  = smax + n;

  init_kernel<<<(n * D_FEAT + 255) / 256, 256, 0, stream>>>(out, smax, denom, n);

  int nblocks = (E + EPB - 1) / EPB;
  edge_mlp_kernel<<<nblocks, 128, 0, stream>>>(x, src, tgt, edge_attr, W1, b1, W2, b2,
                                               av, scores, smax, denom, out, attn_out,
                                               E, 0);
  segmax_kernel<<<(E + 255) / 256, 256, 0, stream>>>(scores, tgt, smax, E);
  segsum_kernel<<<(E + 255) / 256, 256, 0, stream>>>(scores, tgt, smax, denom, E);
  edge_mlp_kernel<<<nblocks, 128, 0, stream>>>(x, src, tgt, edge_attr, W1, b1, W2, b2,
                                               av, scores, smax, denom, out, attn_out,
                                               E, 1);
}